// SememeEmbeddingKNN_70738111365751
// MI455X (gfx1250) — compile-verified
//
#include <hip/hip_runtime.h>
#include <stdint.h>

// Problem shape (from reference): V=50257, H=1024, L=256, W=16, S=32
#define HDIM 1024
#define WWORDS 16
#define SSEM 32

typedef float v2f __attribute__((ext_vector_type(2)));
typedef float v8f __attribute__((ext_vector_type(8)));

// ---- CDNA5 async copy: global -> LDS, 16B per lane, tracked by ASYNCcnt ----
__device__ __forceinline__ void async_load_b128(uint32_t lds_off, const void* gaddr) {
    asm volatile("global_load_async_to_lds_b128 %0, %1, off"
                 :: "v"(lds_off), "v"(gaddr)
                 : "memory");
}
__device__ __forceinline__ void wait_async0() {
    asm volatile("s_wait_asynccnt 0" ::: "memory");
}

__global__ void __launch_bounds__(64)
SememeEmbeddingKNN_70738111365751_kernel(const float* __restrict__ emb,
                                         const int*   __restrict__ word_ids,
                                         const int*   __restrict__ sem_ids,
                                         float*       __restrict__ out) {
    extern __shared__ char smem[];
    float* s_word = (float*)smem;                              // HDIM floats
    float* s_sem  = (float*)(smem + HDIM * 4);                 // SSEM*HDIM floats
    float* s_np   = (float*)(smem + HDIM * 4 + SSEM * HDIM * 4); // 64 norm partials
    float* s_dot  = s_np + 64;                                 // 32 dots
    int*   s_sel  = (int*)(s_dot + 32);                        // 3 selected (pad 4)
    int*   s_idx  = s_sel + 4;                                 // 33 row indices

    const int tid  = threadIdx.x;        // 0..63 (2 waves)
    const int lane = tid & 31;
    const int wv   = tid >> 5;
    const int p    = blockIdx.x;         // flat word slot = l*W + w
    const int l    = p >> 4;

    // Load the 33 embedding-row indices (1 word + 32 sememes) into LDS.
    if (tid < SSEM + 1)
        s_idx[tid] = (tid == 0) ? word_ids[p]
                                : sem_ids[(size_t)p * SSEM + (tid - 1)];
    __syncthreads();

    // ---- Stage word row + 32 sememe rows into LDS (async, coalesced) ----
    const uint32_t off_word = (uint32_t)(uintptr_t)s_word;
    const uint32_t off_sem  = (uint32_t)(uintptr_t)s_sem;
    for (int v = 0; v < SSEM + 1; ++v) {
        const char* grow = (const char*)(emb + (size_t)s_idx[v] * HDIM);
        const uint32_t lrow = (v == 0) ? off_word
                                       : off_sem + (uint32_t)(v - 1) * (HDIM * 4);
        #pragma unroll
        for (int q = 0; q < (HDIM * 4) / (64 * 16); ++q) {   // 4 passes of 1KB
            const uint32_t byt = (uint32_t)(q * 64 + tid) * 16u;
            async_load_b128(lrow + byt, grow + byt);
        }
    }
    wait_async0();
    __syncthreads();

    // ---- Dot products sm_s . w via V_WMMA_F32_16X16X4_F32 ----
    // A (16x4 f32): lanes 0-15 -> M=lane, K={4i,4i+1}; lanes 16-31 -> M=lane-16, K={4i+2,4i+3}
    // B (4x16): word-vector chunk broadcast across all 16 columns (same per-lane K split)
    const int m     = lane & 15;
    const int khalf = lane >> 4;
    const float* srow = s_sem + (size_t)(wv * 16 + m) * HDIM;
    v8f c = {};
    float np = 0.f;
    #pragma unroll 4
    for (int i = 0; i < HDIM / 4; ++i) {
        const int k0 = 4 * i + 2 * khalf;
        const v2f a = *(const v2f*)(srow + k0);
        const v2f b = *(const v2f*)(s_word + k0);
        c = __builtin_amdgcn_wmma_f32_16x16x4_f32(false, a, false, b,
                                                  (short)0, c, false, false);
        np += a.x * a.x + a.y * a.y;   // ||sm||^2 partial, reuses loaded A
    }
    // D layout: VGPR j -> M=j (lanes 0-15) / M=j+8 (lanes 16-31); all N columns equal.
    if (lane == 0) {
        #pragma unroll
        for (int j = 0; j < 8; ++j) s_dot[wv * 16 + j] = c[j];
    } else if (lane == 16) {
        #pragma unroll
        for (int j = 0; j < 8; ++j) s_dot[wv * 16 + 8 + j] = c[j];
    }
    s_np[tid] = np;
    __syncthreads();

    // ---- Top-3 farthest: rank by ||sm||^2 - 2 sm.w (ties: lower index first) ----
    if (tid < SSEM) {
        const int sw = tid >> 4, sm2 = tid & 15;
        const float key = s_np[sw * 32 + sm2] + s_np[sw * 32 + 16 + sm2]
                        - 2.f * s_dot[tid];
        int rank = 0;
        for (int j = 0; j < SSEM; ++j) {
            const float kj = __shfl(key, j, 32);
            rank += (kj > key) || (kj == key && j < tid);
        }
        if (rank < 3) s_sel[rank] = tid;
    }
    __syncthreads();

    // ---- out[l,h] += (1/32)*w[h] + (1/96)*(sum of 3 selected rows) ----
    const int i0 = s_sel[0], i1 = s_sel[1], i2 = s_sel[2];
    float* orow = out + (size_t)l * HDIM;
    #pragma unroll
    for (int q = 0; q < HDIM / 64; ++q) {
        const int h = q * 64 + tid;
        const float v3 = s_sem[(size_t)i0 * HDIM + h]
                       + s_sem[(size_t)i1 * HDIM + h]
                       + s_sem[(size_t)i2 * HDIM + h];
        const float val = (0.5f / WWORDS) * s_word[h]
                        + (0.5f / (3.f * WWORDS)) * v3;
        __hip_atomic_fetch_add(orow + h, val, __ATOMIC_RELAXED,
                               __HIP_MEMORY_SCOPE_AGENT);
    }
}

__global__ void SememeEmbeddingKNN_zero_kernel(float* o, int n) {
    const int i = blockIdx.x * blockDim.x + threadIdx.x;
    if (i < n) o[i] = 0.f;
}

extern "C" void kernel_launch(void* const* d_in, const int* in_sizes, int n_in,
                              void* d_out, int out_size, void* d_ws, size_t ws_size,
                              hipStream_t stream) {
    const float* emb = (const float*)d_in[0];   // [V,H] f32
    const int*   wid = (const int*)d_in[1];     // [L,W] i32
    const int*   sid = (const int*)d_in[2];     // [L,W,S] i32
    float* o = (float*)d_out;                   // [L,H] f32

    const int n_words = in_sizes[1];            // L*W = 4096

    SememeEmbeddingKNN_zero_kernel<<<(out_size + 255) / 256, 256, 0, stream>>>(o, out_size);

    const size_t shmem = (size_t)(SSEM + 1) * HDIM * 4   // staged rows
                       + 64 * 4 + 32 * 4 + 4 * 4 + 36 * 4; // partials/dots/sel/idx
    SememeEmbeddingKNN_70738111365751_kernel<<<n_words, 64, shmem, stream>>>(emb, wid, sid, o);
}